// GATLayer_1580547972217
// MI455X (gfx1250) — compile-verified
//
#include <hip/hip_runtime.h>

typedef __attribute__((ext_vector_type(2))) float v2f;
typedef __attribute__((ext_vector_type(4))) float v4f;
typedef __attribute__((ext_vector_type(8))) float v8f;

#define N_NODES 2048
#define IN_F    128
#define OUT_F   64
#define NEG_INF (-9.0e15f)
#define PSTR    2052   // 16 x 2048 score block, padded: 2052 % 64 == 4 -> no bank collisions

// ---------------------------------------------------------------------------
// Kernel A: h = x @ W   ([2048,128] x [128,64])  via V_WMMA_F32_16X16X4_F32
// 512 tiles of 16x16, one wave per tile, 8 waves/block -> 64 blocks.
// ---------------------------------------------------------------------------
__global__ __launch_bounds__(256) void gat_xw_wmma(
    const float* __restrict__ x, const float* __restrict__ W,
    float* __restrict__ h) {
  const int tid  = threadIdx.x;
  const int lane = tid & 31;
  const int wave = tid >> 5;
  const int tile = blockIdx.x * 8 + wave;      // 0..511
  const int rowT = tile >> 2;                  // 0..127
  const int colT = tile & 3;                   // 0..3

  const int m    = rowT * 16 + (lane & 15);    // A: M = lane&15
  const int n    = colT * 16 + (lane & 15);    // B/D: N = lane&15
  const int koff = (lane >> 4) * 2;            // lanes 16..31 hold K+2,K+3

  v8f acc = {};
#pragma unroll
  for (int k = 0; k < IN_F; k += 4) {
    v2f av = *(const v2f*)&x[m * IN_F + k + koff];          // K, K+1 contiguous
    v2f bv;
    bv.x = W[(k + koff) * OUT_F + n];
    bv.y = W[(k + koff + 1) * OUT_F + n];
    acc = __builtin_amdgcn_wmma_f32_16x16x4_f32(
        /*neg_a=*/false, av, /*neg_b=*/false, bv,
        /*c_mod=*/(short)0, acc, /*reuse_a=*/false, /*reuse_b=*/false);
  }
  const int rbase = rowT * 16 + ((lane >> 4) << 3);         // M = r + 8*(lane>>4)
#pragma unroll
  for (int r = 0; r < 8; ++r) h[(rbase + r) * OUT_F + n] = acc[r];
}

// ---------------------------------------------------------------------------
// Kernel B: fused scores + masked softmax + attention@h + relu.
// One block per 16-row block. 256 threads = 8 wave32.
// LDS: hi[16*64] | a[64] | P[16*2052] | scr[4*256]   = 34944 floats = 139776 B
// ---------------------------------------------------------------------------
__global__ __launch_bounds__(256) void gat_fused(
    const float* __restrict__ h, const int* __restrict__ adj,
    const float* __restrict__ a, float* __restrict__ out) {
  extern __shared__ float smem[];
  float* hiS = smem;                    // 16*64
  float* aS  = smem + 16 * OUT_F;       // 64
  float* Ps  = aS + OUT_F;              // 16*PSTR
  float* scr = Ps + 16 * PSTR;          // 4*256

  const int tid      = threadIdx.x;
  const int lane     = tid & 31;
  const int wave     = tid >> 5;
  const int blockRow = blockIdx.x;      // 0..127
  const int i0       = blockRow * 16;

  // ---- Phase 1: stage h_i block and attention vector a into LDS ----
  for (int idx = tid; idx < 16 * OUT_F; idx += 256)
    hiS[idx] = h[(i0 + (idx >> 6)) * OUT_F + (idx & 63)];
  if (tid < OUT_F) aS[tid] = a[tid];
  __syncthreads();

  // ---- Phase 2: raw masked scores into LDS P (thread owns column j) ----
  for (int j = tid; j < N_NODES; j += 256) {
    __builtin_prefetch(&h[(j + 256 < N_NODES ? j + 256 : j) * OUT_F], 0, 0);
    v4f hj[16];
#pragma unroll
    for (int q = 0; q < 16; ++q) hj[q] = *(const v4f*)&h[j * OUT_F + q * 4];
    for (int i = 0; i < 16; ++i) {
      float e = 0.0f;
#pragma unroll
      for (int q = 0; q < 16; ++q) {
        v4f hi4 = *(const v4f*)&hiS[i * OUT_F + q * 4];   // LDS broadcast
        v4f a4  = *(const v4f*)&aS[q * 4];                // LDS broadcast
        e += __builtin_fabsf(hj[q].x - hi4.x) * a4.x;
        e += __builtin_fabsf(hj[q].y - hi4.y) * a4.y;
        e += __builtin_fabsf(hj[q].z - hi4.z) * a4.z;
        e += __builtin_fabsf(hj[q].w - hi4.w) * a4.w;
      }
      e = fmaxf(e, 0.0f);                                 // relu
      const int ad = adj[(i0 + i) * N_NODES + j];         // coalesced over tid
      Ps[i * PSTR + j] = (ad > 0) ? e : NEG_INF;
    }
  }
  __syncthreads();

  // ---- Phase 3: row-wise softmax in LDS (wave owns 2 rows) ----
  for (int rr = 0; rr < 2; ++rr) {
    const int i = wave * 2 + rr;
    float m = NEG_INF;
    for (int j = lane; j < N_NODES; j += 32) m = fmaxf(m, Ps[i * PSTR + j]);
#pragma unroll
    for (int off = 16; off > 0; off >>= 1) m = fmaxf(m, __shfl_xor(m, off, 32));
    float s = 0.0f;
    for (int j = lane; j < N_NODES; j += 32) {
      float t = __expf(Ps[i * PSTR + j] - m);   // exp(-9e15 - m) == 0 -> mask
      Ps[i * PSTR + j] = t;
      s += t;
    }
#pragma unroll
    for (int off = 16; off > 0; off >>= 1) s += __shfl_xor(s, off, 32);
    const float inv = 1.0f / s;
    for (int j = lane; j < N_NODES; j += 32) Ps[i * PSTR + j] *= inv;
  }
  __syncthreads();

  // ---- Phase 4: out_block = P @ h via WMMA f32 16x16x4 ----
  // wave&3 -> feature tile (0..3); wave>>2 -> K half (0..1)
  const int nt    = wave & 3;
  const int kh    = wave >> 2;
  const int col   = nt * 16 + (lane & 15);
  const int koff  = (lane >> 4) * 2;
  const int arow  = lane & 15;          // A-matrix M index

  v8f acc = {};
  const int kBeg = kh * (N_NODES / 2), kEnd = kBeg + N_NODES / 2;
  for (int k = kBeg; k < kEnd; k += 4) {
    v2f av = *(const v2f*)&Ps[arow * PSTR + k + koff];   // ds_load_b64, no conflicts
    v2f bv;
    bv.x = h[(k + koff) * OUT_F + col];
    bv.y = h[(k + koff + 1) * OUT_F + col];
    acc = __builtin_amdgcn_wmma_f32_16x16x4_f32(
        false, av, false, bv, (short)0, acc, false, false);
  }

  // combine the two K-halves through LDS, relu, store
  if (kh == 1) {
#pragma unroll
    for (int r = 0; r < 8; ++r) scr[nt * 256 + r * 32 + lane] = acc[r];
  }
  __syncthreads();
  if (kh == 0) {
    const int rbase = (lane >> 4) << 3;                  // M = r + 8*(lane>>4)
#pragma unroll
    for (int r = 0; r < 8; ++r) {
      float v = acc[r] + scr[nt * 256 + r * 32 + lane];
      out[(i0 + rbase + r) * OUT_F + col] = fmaxf(v, 0.0f);
    }
  }
}

// ---------------------------------------------------------------------------
extern "C" void kernel_launch(void* const* d_in, const int* in_sizes, int n_in,
                              void* d_out, int out_size, void* d_ws, size_t ws_size,
                              hipStream_t stream) {
  const float* x   = (const float*)d_in[0];   // [2048,128] f32
  const int*   adj = (const int*)d_in[1];     // [2048,2048] i32
  const float* W   = (const float*)d_in[2];   // [128,64] f32
  const float* a   = (const float*)d_in[3];   // [64,1] f32
  float*       out = (float*)d_out;           // [2048,64] f32
  float*       h   = (float*)d_ws;            // 2048*64*4 = 512 KB scratch

  // h = x @ W
  gat_xw_wmma<<<64, 256, 0, stream>>>(x, W, h);

  // fused scores/softmax/attention GEMM: one block per 16-row block
  const size_t smemBytes =
      (size_t)(16 * OUT_F + OUT_F + 16 * PSTR + 4 * 256) * sizeof(float); // 139776 B
  gat_fused<<<N_NODES / 16, 256, smemBytes, stream>>>(h, adj, a, out);
}